// Patcher_58634893525578
// MI455X (gfx1250) — compile-verified
//
#include <hip/hip_runtime.h>

#define NB 16          // batch
#define NN 16          // boxes per image
#define IH 640
#define IW 640
#define HP 300
#define WP 300
#define IMG_ELEMS (NB * IH * IW * 3)

struct BoxRec { int y0, x0, s, flags; }; // flags: bit0=rot90, bit1=flip_lr, bit2=flip_ud, bit3=valid

// ---------------------------------------------------------------------------
// Kernel A: compute float patch boxes (output #2) and packed integer box
// records (workspace) exactly like Patcher.create() + the int cast/area gate.
// ---------------------------------------------------------------------------
__global__ __launch_bounds__(256) void prep_boxes(
    const float* __restrict__ boxes,     // [B,N,4] ymin,xmin,h,w
    const int*   __restrict__ dec,       // [B,N,3] raw 32-bit words; !=0 test
    float*       __restrict__ pboxes,    // [B,N,4] -> d_out tail
    BoxRec*      __restrict__ recs)      // [B*N]   -> workspace
{
    int i = blockIdx.x * blockDim.x + threadIdx.x;
    if (i >= NB * NN) return;
    float ymin = boxes[i * 4 + 0];
    float xmin = boxes[i * 4 + 1];
    float h    = boxes[i * 4 + 2];
    float w    = boxes[i * 4 + 3];
    float ph = h * 0.2f;                 // SCALE
    float pw = ph;                       // ASPECT == 1
    float y0 = ymin + h * 0.5f;          // ORIGIN
    float x0 = xmin + w * 0.5f;
    if (y0 + ph > (float)IH) y0 = (float)IH - ph;
    if (x0 + pw > (float)IW) x0 = (float)IW - pw;
    pboxes[i * 4 + 0] = y0;
    pboxes[i * 4 + 1] = x0;
    pboxes[i * 4 + 2] = ph;
    pboxes[i * 4 + 3] = pw;

    int yi = (int)y0, xi = (int)x0, hi = (int)ph, wi = (int)pw; // trunc like tf.cast
    int valid = (hi * wi) > 400;
    int s = hi < 1 ? 1 : hi;             // jnp.maximum(box[2], 1)
    int f = (dec[i * 3 + 0] != 0 ? 1 : 0)
          | (dec[i * 3 + 1] != 0 ? 2 : 0)
          | (dec[i * 3 + 2] != 0 ? 4 : 0)
          | (valid ? 8 : 0);
    BoxRec r; r.y0 = yi; r.x0 = xi; r.s = s; r.flags = f;
    recs[i] = r;
}

// ---------------------------------------------------------------------------
// Clamped bilinear sample of the 300x300x3 patch (half-pixel-center map done
// by caller). Matches _bilinear_sample exactly.
// ---------------------------------------------------------------------------
__device__ __forceinline__ void sample_patch(const float* __restrict__ patch,
                                             float sy, float sx, float* rgb)
{
    sy = fminf(fmaxf(sy, 0.0f), (float)(HP - 1));
    sx = fminf(fmaxf(sx, 0.0f), (float)(WP - 1));
    float fy = floorf(sy), fx = floorf(sx);
    int y0 = (int)fy, x0 = (int)fx;
    int y1 = y0 + 1 < HP - 1 ? y0 + 1 : HP - 1;
    int x1 = x0 + 1 < WP - 1 ? x0 + 1 : WP - 1;
    float wy = sy - fy, wx = sx - fx;
    const float* p00 = patch + (size_t)(y0 * WP + x0) * 3;
    const float* p01 = patch + (size_t)(y0 * WP + x1) * 3;
    const float* p10 = patch + (size_t)(y1 * WP + x0) * 3;
    const float* p11 = patch + (size_t)(y1 * WP + x1) * 3;
#pragma unroll
    for (int c = 0; c < 3; ++c) {
        float top = p00[c] * (1.0f - wx) + p01[c] * wx;
        float bot = p10[c] * (1.0f - wx) + p11[c] * wx;
        rgb[c] = top * (1.0f - wy) + bot * wy;
    }
}

// ---------------------------------------------------------------------------
// Kernel B: one thread = one 4-pixel quad (12 floats = 3x b128 load/store).
// Each 256-thread block lies entirely inside one image (102400 quads/image,
// 400 blocks/image), so box records are staged once into LDS.
// Last-write-wins == highest box index covering the pixel.
// ---------------------------------------------------------------------------
__global__ __launch_bounds__(256) void apply_patches(
    const float*  __restrict__ img,     // [B,H,W,3]
    const float*  __restrict__ patch,   // [300,300,3]
    const BoxRec* __restrict__ recs,    // [B*N]
    float*        __restrict__ out)     // [B,H,W,3]
{
    const int quadsPerRow   = IW / 4;                       // 160
    const int blocksPerImg  = (IH * quadsPerRow) / 256;     // 400
    int b     = blockIdx.x / blocksPerImg;
    int local = (blockIdx.x % blocksPerImg) * 256 + threadIdx.x;
    int y     = local / quadsPerRow;
    int xq    = (local % quadsPerRow) * 4;

    // Warm the whole patch (1.08 MB) into L2 via gfx1250 global_prefetch_b8.
    // 33 blocks * 256 threads * 128B lines covers 300*300*3*4 bytes.
    if (blockIdx.x < 33) {
        size_t off = ((size_t)(blockIdx.x * 256 + threadIdx.x)) * 32; // floats (128B)
        if (off < (size_t)HP * WP * 3)
            __builtin_prefetch(patch + off, 0, 0);
    }

    __shared__ BoxRec sb[NN];
    if (threadIdx.x < NN) sb[threadIdx.x] = recs[b * NN + threadIdx.x];
    __syncthreads();

    size_t base = ((size_t)(b * IH + y) * IW + xq) * 3;
    float4 v0 = *(const float4*)(img + base);
    float4 v1 = *(const float4*)(img + base + 4);
    float4 v2 = *(const float4*)(img + base + 8);
    float px[12] = { v0.x, v0.y, v0.z, v0.w,
                     v1.x, v1.y, v1.z, v1.w,
                     v2.x, v2.y, v2.z, v2.w };

    // Winner search: y-test hoisted per box, shared across the quad.
    int win[4] = { -1, -1, -1, -1 };
#pragma unroll
    for (int n = NN - 1; n >= 0; --n) {
        BoxRec r = sb[n];
        int iy = y - r.y0;
        bool vy = (r.flags & 8) && (iy >= 0) && (iy < r.s);
#pragma unroll
        for (int p = 0; p < 4; ++p) {
            int ix = xq + p - r.x0;
            if (win[p] < 0 && vy && ix >= 0 && ix < r.s) win[p] = n;
        }
    }

#pragma unroll
    for (int p = 0; p < 4; ++p) {
        if (win[p] < 0) continue;
        BoxRec r = sb[win[p]];
        int s  = r.s;
        int iy = y - r.y0;
        int ix = xq + p - r.x0;            // inside => already in [0, s)
        int iy2 = (r.flags & 4) ? (s - 1 - iy) : iy;   // invert flip_up_down
        int ix2 = (r.flags & 2) ? (s - 1 - ix) : ix;   // invert flip_left_right
        int ry  = (r.flags & 1) ? ix2 : iy2;           // invert rot90 (CCW)
        int rx  = (r.flags & 1) ? (s - 1 - iy2) : ix2;
        float sf = (float)s;
        float k  = 300.0f / sf;                        // Hp/sf == Wp/sf
        float sy = ((float)ry + 0.5f) * k - 0.5f;
        float sx = ((float)rx + 0.5f) * k - 0.5f;
        sample_patch(patch, sy, sx, &px[p * 3]);
    }

    *(float4*)(out + base)     = make_float4(px[0], px[1], px[2],  px[3]);
    *(float4*)(out + base + 4) = make_float4(px[4], px[5], px[6],  px[7]);
    *(float4*)(out + base + 8) = make_float4(px[8], px[9], px[10], px[11]);
}

extern "C" void kernel_launch(void* const* d_in, const int* in_sizes, int n_in,
                              void* d_out, int out_size, void* d_ws, size_t ws_size,
                              hipStream_t stream) {
    const float* boxes  = (const float*)d_in[0];   // [16,16,4]
    const float* images = (const float*)d_in[1];   // [16,640,640,3]
    const float* patch  = (const float*)d_in[2];   // [300,300,3]
    const int*   dec    = (const int*)d_in[3];     // [16,16,3] bool-as-words

    float* out_img    = (float*)d_out;             // first output, flat
    float* out_pboxes = (float*)d_out + IMG_ELEMS; // second output, flat
    BoxRec* recs      = (BoxRec*)d_ws;             // 4 KB scratch

    prep_boxes<<<1, 256, 0, stream>>>(boxes, dec, out_pboxes, recs);

    const int totalQuads = NB * IH * (IW / 4);     // 1,638,400
    apply_patches<<<totalQuads / 256, 256, 0, stream>>>(images, patch, recs, out_img);
}